// MLPMessageAggregator_60318520705540
// MI455X (gfx1250) — compile-verified
//
#include <hip/hip_runtime.h>

typedef __attribute__((ext_vector_type(2))) float v2f;
typedef __attribute__((ext_vector_type(8))) float v8f;

#define K_SLOTS 6
#define MWID    688
#define NTILES  43    // 688 / 16 exactly

// One block per node. 8 waves per block; each wave owns m-tiles t, t+8, ...
// Layer-1 (6->3) done on the matrix pipe:
//   D(16x16) = A(16x4) x B(4x16)  chained twice over K (k0..3, k4..5+pad)
//   A = counts-masked W1^T (rows = hidden units j, padded 3->16)
//   B = 16 consecutive m columns of msgs for node n
// D rows 0..2 (VGPRs acc[0..2], lanes 0..15, col m = lane) feed the scalar
// layer-2/3 epilogue in-lane. Rows 3..15 are exact zeros (A rows zero, C=0).
__global__ __launch_bounds__(256)
void mlp_msg_agg_kernel(const float* __restrict__ msgs,
                        const int*   __restrict__ counts,
                        const float* __restrict__ ts,
                        const float* __restrict__ W1,
                        const float* __restrict__ b1,
                        const float* __restrict__ W2,
                        const float* __restrict__ b2,
                        const float* __restrict__ W3,
                        const float* __restrict__ b3,
                        float*       __restrict__ out,
                        float*       __restrict__ out_ts)
{
    const int n    = blockIdx.x;
    const int tid  = threadIdx.x;
    const int lane = tid & 31;
    const int wave = tid >> 5;
    const int half = lane >> 4;   // lane half: selects K rows per ISA layout
    const int j    = lane & 15;   // A: hidden-unit row; B/D: m within tile

    if (tid == 0) out_ts[n] = ts[n];   // timestamps passthrough

    // Per-node mask folded into the weights: slot k valid iff k >= 6 - count.
    const int kmin = K_SLOTS - counts[n];

    // A operands (ISA 7.12.2, 32-bit A 16x4): lanes 0-15 -> VGPR0=K0,VGPR1=K1;
    // lanes 16-31 -> VGPR0=K2,VGPR1=K3. First WMMA (A2) covers k=4,5 (rows 2,3 = 0).
    float a1x = 0.f, a1y = 0.f, a2x = 0.f, a2y = 0.f;
    if (j < 3) {
        const int k0 = 2 * half;
        a1x = (k0     >= kmin) ? W1[ k0      * 3 + j] : 0.f;
        a1y = (k0 + 1 >= kmin) ? W1[(k0 + 1) * 3 + j] : 0.f;
        if (half == 0) {
            a2x = (4 >= kmin) ? W1[4 * 3 + j] : 0.f;
            a2y = (5 >= kmin) ? W1[5 * 3 + j] : 0.f;
        }
    }
    const v2f A1 = { a1x, a1y };
    const v2f A2 = { a2x, a2y };

    // Tiny epilogue weights (hit constant-cached lines; negligible traffic).
    const float B1_0 = b1[0], B1_1 = b1[1], B1_2 = b1[2];
    const float w200 = W2[0], w201 = W2[1], w210 = W2[2],
                w211 = W2[3], w220 = W2[4], w221 = W2[5];
    const float B2_0 = b2[0], B2_1 = b2[1];
    const float w30  = W3[0], w31  = W3[1], B3 = b3[0];

    const float* __restrict__ base = msgs + (size_t)n * (K_SLOTS * MWID);
    float*       __restrict__ orow = out  + (size_t)n * MWID;

    const int kr = 2 * half;   // B rows: lanes 0-15 -> k=0,1 ; lanes 16-31 -> k=2,3

    for (int t = wave; t < NTILES; t += 8) {
        const int m0 = t * 16;

        // B operand: one 16-float row segment per (VGPR, lane-half).
        v2f Bv1, Bv2;
        Bv1.x = __builtin_nontemporal_load(base + (kr    ) * MWID + m0 + j);
        Bv1.y = __builtin_nontemporal_load(base + (kr + 1) * MWID + m0 + j);
        if (half == 0) {
            Bv2.x = __builtin_nontemporal_load(base + 4 * MWID + m0 + j);
            Bv2.y = __builtin_nontemporal_load(base + 5 * MWID + m0 + j);
        } else {
            Bv2.x = 0.f; Bv2.y = 0.f;   // padded K rows 6,7
        }

        // Prefetch this wave's next tile (stride 8 tiles = 128 floats).
        if (t + 8 < NTILES)
            __builtin_prefetch(base + kr * MWID + m0 + 128 + j, 0, 0);

        // Chained K accumulation on the matrix pipe (EXEC all-1s here).
        v8f acc = {};
        acc = __builtin_amdgcn_wmma_f32_16x16x4_f32(
                  false, A2, false, Bv2, (short)0, acc, false, false);
        acc = __builtin_amdgcn_wmma_f32_16x16x4_f32(
                  false, A1, false, Bv1, (short)0, acc, false, false);

        // Lanes 0-15 hold h1_pre[j_hidden][m=lane] in acc[0..2]; finish MLP.
        if (half == 0) {
            const float h0 = fmaxf(acc[0] + B1_0, 0.f);
            const float h1 = fmaxf(acc[1] + B1_1, 0.f);
            const float h2 = fmaxf(acc[2] + B1_2, 0.f);
            const float g0 = fmaxf(h0 * w200 + h1 * w210 + h2 * w220 + B2_0, 0.f);
            const float g1 = fmaxf(h0 * w201 + h1 * w211 + h2 * w221 + B2_1, 0.f);
            __builtin_nontemporal_store(g0 * w30 + g1 * w31 + B3, orow + m0 + j);
        }
    }
}

extern "C" void kernel_launch(void* const* d_in, const int* in_sizes, int n_in,
                              void* d_out, int out_size, void* d_ws, size_t ws_size,
                              hipStream_t stream) {
    const float* msgs   = (const float*)d_in[0];
    const int*   counts = (const int*)  d_in[1];
    const float* ts     = (const float*)d_in[2];
    const float* W1     = (const float*)d_in[3];
    const float* b1     = (const float*)d_in[4];
    const float* W2     = (const float*)d_in[5];
    const float* b2     = (const float*)d_in[6];
    const float* W3     = (const float*)d_in[7];
    const float* b3     = (const float*)d_in[8];

    const int N = in_sizes[1];                 // counts has N elements
    float* out    = (float*)d_out;             // [N, 688]
    float* out_ts = out + (size_t)N * MWID;    // then [N] timestamps

    mlp_msg_agg_kernel<<<N, 256, 0, stream>>>(msgs, counts, ts,
                                              W1, b1, W2, b2, W3, b3,
                                              out, out_ts);
}